// Decoding_43404939493634
// MI455X (gfx1250) — compile-verified
//
#include <hip/hip_runtime.h>

typedef __attribute__((ext_vector_type(2))) float v2f;
typedef __attribute__((ext_vector_type(4))) float v4f;
typedef __attribute__((ext_vector_type(8))) float v8f;

#define EMB 64

// Select element j (runtime 0..7) from a v8f with constant-index cndmask chain
// (avoids dynamic VGPR indexing / scratch spills).
__device__ __forceinline__ float sel8(v8f a, unsigned j) {
    float r = a[0];
    r = (j == 1u) ? a[1] : r;
    r = (j == 2u) ? a[2] : r;
    r = (j == 3u) ? a[3] : r;
    r = (j == 4u) ? a[4] : r;
    r = (j == 5u) ? a[5] : r;
    r = (j == 6u) ? a[6] : r;
    r = (j == 7u) ? a[7] : r;
    return r;
}

// Pass 1: per node n, P[n] = sum_e max(z,0)*W5[e], M[n] = sum_e min(z,0)*W5[e].
// Streams the 51.2MB node table once (non-temporal); PM table (1.6MB) stays hot.
__global__ void precompute_pm(const float* __restrict__ node,
                              const float* __restrict__ W5,
                              float2* __restrict__ pm, int n_nodes) {
    int n = blockIdx.x * blockDim.x + threadIdx.x;
    if (n >= n_nodes) return;
    const float* row = node + (size_t)n * EMB;
    float p = 0.f, m = 0.f;
#pragma unroll
    for (int e = 0; e < EMB; e += 4) {
        v4f z = __builtin_nontemporal_load((const v4f*)(row + e));
        v4f w = *(const v4f*)(W5 + e);
#pragma unroll
        for (int t = 0; t < 4; ++t) {
            p = fmaf(fmaxf(z[t], 0.f), w[t], p);
            m = fmaf(fminf(z[t], 0.f), w[t], m);
        }
    }
    pm[n] = make_float2(p, m);
}

// Pass 2: per wave, 32 rows of s = state @ W4 via V_WMMA_F32_16X16X4_F32,
// then Q = s * (s>=0 ? P[idx] : M[idx]).
__global__ void qsa_wmma(const int* __restrict__ idx,
                         const float* __restrict__ state,
                         const float* __restrict__ W4,
                         const float2* __restrict__ pm,
                         float* __restrict__ out, int batch) {
    const unsigned l = threadIdx.x & 31u;
    const long wave = (long)((blockIdx.x * blockDim.x + threadIdx.x) >> 5);
    const long rowbase = wave * 32;
    if (rowbase >= batch) return;                 // wave-uniform

    const unsigned koff = (l >> 4) << 1;          // lanes 0-15: K base+0/1; 16-31: K base+2/3

    if (rowbase + 32 <= (long)batch) {            // wave-uniform: EXEC all-ones for WMMA
        // B operand: W4 slice replicated across all 16 N columns -> every D column
        // holds the same 16 dot products. 16 float2 = 32 VGPRs, loaded once.
        v2f b[16];
#pragma unroll
        for (int k = 0; k < 16; ++k)
            b[k] = *(const v2f*)(W4 + k * 4 + koff);

        // A operand layout (32-bit A 16x4): lane m & lane m+16 share row m,
        // holding K = koff, koff+1 in the two VGPRs.
        const float* a0p = state + (size_t)(rowbase + (l & 15u)) * EMB + koff;
        const float* a1p = a0p + 16 * EMB;

        v8f acc0 = {};  // rows rowbase+0..15
        v8f acc1 = {};  // rows rowbase+16..31
#pragma unroll
        for (int k = 0; k < 16; ++k) {
            v2f a0 = __builtin_nontemporal_load((const v2f*)(a0p + k * 4));
            v2f a1 = __builtin_nontemporal_load((const v2f*)(a1p + k * 4));
            acc0 = __builtin_amdgcn_wmma_f32_16x16x4_f32(
                false, a0, false, b[k], (short)0, acc0, false, false);
            acc1 = __builtin_amdgcn_wmma_f32_16x16x4_f32(
                false, a1, false, b[k], (short)0, acc1, false, false);
        }

        // C/D layout: VGPR j = row j (lanes 0-15) / row 8+j (lanes 16-31).
        // All N columns identical => lane l owns tile-row r = lane id with bits 3<->4 swapped.
        unsigned j = l & 7u;
        unsigned r = j | ((l & 8u) << 1) | ((l & 16u) >> 1);
        float s = (l & 8u) ? sel8(acc1, j) : sel8(acc0, j);

        long row = rowbase + (long)r;
        int id = __builtin_nontemporal_load(idx + row);
        float2 pmv = pm[id];                      // hot 1.6MB table, cached
        float q = s * (s >= 0.f ? pmv.x : pmv.y);
        __builtin_nontemporal_store(q, out + row);
    } else {
        // Remainder wave (not hit for B=400000): scalar per-lane path.
        long row = rowbase + (long)l;
        if (row < (long)batch) {
            const float* sr = state + (size_t)row * EMB;
            float s = 0.f;
#pragma unroll
            for (int e = 0; e < EMB; ++e) s = fmaf(sr[e], W4[e], s);
            int id = idx[row];
            float2 pmv = pm[id];
            out[row] = s * (s >= 0.f ? pmv.x : pmv.y);
        }
    }
}

// Fallback single-pass kernel if workspace can't hold the PM table.
__global__ void qsa_direct(const int* __restrict__ idx,
                           const float* __restrict__ node,
                           const float* __restrict__ state,
                           const float* __restrict__ W4,
                           const float* __restrict__ W5,
                           float* __restrict__ out, int batch) {
    long b = (long)blockIdx.x * blockDim.x + threadIdx.x;
    if (b >= batch) return;
    const float* sr = state + (size_t)b * EMB;
    float s = 0.f;
#pragma unroll
    for (int e = 0; e < EMB; ++e) s = fmaf(sr[e], W4[e], s);
    const float* zr = node + (size_t)idx[b] * EMB;
    float q = 0.f;
#pragma unroll
    for (int e = 0; e < EMB; ++e) q = fmaf(fmaxf(zr[e] * s, 0.f), W5[e], q);
    out[b] = q;
}

extern "C" void kernel_launch(void* const* d_in, const int* in_sizes, int n_in,
                              void* d_out, int out_size, void* d_ws, size_t ws_size,
                              hipStream_t stream) {
    const int*   actions = (const int*)d_in[0];
    const float* node    = (const float*)d_in[1];
    const float* state   = (const float*)d_in[2];
    const float* W4      = (const float*)d_in[3];
    const float* W5      = (const float*)d_in[4];
    float*       out     = (float*)d_out;

    int batch   = in_sizes[0];
    int n_nodes = in_sizes[1] / EMB;

    size_t pm_bytes = (size_t)n_nodes * sizeof(float2);
    if (ws_size >= pm_bytes) {
        float2* pm = (float2*)d_ws;
        precompute_pm<<<(n_nodes + 255) / 256, 256, 0, stream>>>(node, W5, pm, n_nodes);
        long waves = ((long)batch + 31) / 32;
        long threads = waves * 32;
        qsa_wmma<<<(unsigned)((threads + 255) / 256), 256, 0, stream>>>(
            actions, state, W4, pm, out, batch);
    } else {
        qsa_direct<<<(batch + 255) / 256, 256, 0, stream>>>(
            actions, node, state, W4, W5, out, batch);
    }
}